// ManualConv2d_59622736003688
// MI455X (gfx1250) — compile-verified
//
#include <hip/hip_runtime.h>

typedef float v2f __attribute__((ext_vector_type(2)));
typedef float v8f __attribute__((ext_vector_type(8)));

// Problem constants (from reference)
constexpr int N_   = 16;
constexpr int IC   = 64;
constexpr int OC   = 128;
constexpr int H_   = 112;
constexpr int W_   = 112;
constexpr int HW   = H_ * W_;          // 12544
constexpr int NSP  = N_ * HW;          // 200704 (GEMM N)
constexpr int KG   = IC * 9;           // 576   (GEMM K)

// Tiling
constexpr int KB     = 32;             // K chunk staged in LDS
constexpr int KH     = KB / 2;         // K-pair rows (v2f-packed along K)
constexpr int NCHUNK = KG / KB;        // 18
constexpr int SPB    = 128;            // spatial tile per block (GEMM N tile)
// v2f pitch 144 -> 288 dwords -> row shift = 288 mod 64 = 32 banks:
// the two K-pair rows read by one fragment land in disjoint bank halves.
constexpr int PW     = OC + 16;        // 144
constexpr int PI     = SPB + 16;       // 144

__global__ __launch_bounds__(256) void conv3x3_wmma_f32(
    const float* __restrict__ x,      // [N, IC, H, W]
    const float* __restrict__ wgt,    // [OC, IC, 3, 3] == [OC, KG]
    const float* __restrict__ bias,   // [OC]
    float* __restrict__ out)          // [N, OC, H, W]
{
    __shared__ v2f wT[KH][PW];         // [k/2][oc]  = {w(k), w(k+1)}
    __shared__ v2f iT[KH][PI];         // [k/2][sp]  = {x(k), x(k+1)} (im2col)

    const int tid  = threadIdx.x;
    const int wave = tid >> 5;             // 0..7
    const int lane = tid & 31;

    const int oc0     = (wave & 3) * 32;   // 2 oc tiles: oc0, oc0+16
    const int spw     = (wave >> 2) * 64;  // 4 sp tiles: spw + {0,16,32,48}
    const int sp_base = blockIdx.x * SPB;  // exact cover: 1568 blocks

    // Fragment coordinates (ISA 7.12.2, 32-bit A/B layouts)
    const int fm  = lane & 15;             // M for A, N for B
    const int fkp = lane >> 4;             // K-pair row offset: 0 or 1 (K off 0 / 2)

    // ---- staging coordinates (loop-invariant per thread) ----
    // weights: one oc row, 16 consecutive k -> b128 global loads, b64 LDS stores
    const int oc_w    = tid >> 1;          // 0..127
    const int kbase_w = (tid & 1) * 16;    // 0 or 16
    // inputs: one fixed spatial position, k = kb_i + 2j -> coalesced along iw
    const int sp_i  = tid & 127;
    const int kb_i  = tid >> 7;            // 0 or 1 (component within v2f)
    const int gsp_i = sp_base + sp_i;
    const int n_i   = gsp_i / HW;
    const int rem_i = gsp_i - n_i * HW;
    const int oh_i  = rem_i / W_;
    const int ow_i  = rem_i - oh_i * W_;

    float wreg[16];
    float ireg[16];

    auto prefetch = [&](int k0) {
        const float* __restrict__ wp = wgt + (oc_w * KG + k0 + kbase_w);
        #pragma unroll
        for (int j = 0; j < 16; ++j) wreg[j] = wp[j];
        #pragma unroll
        for (int j = 0; j < 16; ++j) {
            const int k  = k0 + kb_i + 2 * j;
            const int ic = k / 9;
            const int r  = k - ic * 9;
            const int kh = r / 3;
            const int kw = r - kh * 3;
            const int ih = oh_i + kh - 1;
            const int iw = ow_i + kw - 1;
            float v = 0.0f;
            if ((unsigned)ih < (unsigned)H_ && (unsigned)iw < (unsigned)W_)
                v = x[((n_i * IC + ic) * H_ + ih) * W_ + iw];
            ireg[j] = v;
        }
    };

    v8f acc[2][4] = {};

    prefetch(0);
    for (int c = 0; c < NCHUNK; ++c) {
        __syncthreads();                   // previous chunk fully consumed
        // weights: 8 x ds_store_b64 (K pairs packed)
        #pragma unroll
        for (int jp = 0; jp < 8; ++jp) {
            v2f p; p.x = wreg[2 * jp]; p.y = wreg[2 * jp + 1];
            wT[(kbase_w >> 1) + jp][oc_w] = p;
        }
        // inputs: this thread owns one component (kb_i) of each K pair
        #pragma unroll
        for (int j = 0; j < 16; ++j)
            ((float*)&iT[j][sp_i])[kb_i] = ireg[j];
        __syncthreads();                   // staging visible
        if (c + 1 < NCHUNK) prefetch((c + 1) * KB);  // overlap with WMMAs below

        #pragma unroll
        for (int kk = 0; kk < KB; kk += 4) {
            const int kp = (kk >> 1) + fkp;        // K-pair row for this lane half
            v2f a0, a1, b[4];
            a0 = wT[kp][oc0 + fm];                 // single ds_load_b64 each
            a1 = wT[kp][oc0 + 16 + fm];
            #pragma unroll
            for (int s = 0; s < 4; ++s)
                b[s] = iT[kp][spw + s * 16 + fm];
            #pragma unroll
            for (int s = 0; s < 4; ++s) {
                acc[0][s] = __builtin_amdgcn_wmma_f32_16x16x4_f32(
                    false, a0, false, b[s], (short)0, acc[0][s], false, false);
                acc[1][s] = __builtin_amdgcn_wmma_f32_16x16x4_f32(
                    false, a1, false, b[s], (short)0, acc[1][s], false, false);
            }
        }
    }

    // ---- writeback: D layout -> oc = base + i + 8*(lane>=16), sp col = lane&15 ----
    const int ochalf = (lane >> 4) * 8;
    #pragma unroll
    for (int s = 0; s < 4; ++s) {
        const int gsp = sp_base + spw + s * 16 + fm;
        const int n   = gsp / HW;
        const int rem = gsp - n * HW;
        #pragma unroll
        for (int t = 0; t < 2; ++t) {
            #pragma unroll
            for (int i = 0; i < 8; ++i) {
                const int oc = oc0 + t * 16 + ochalf + i;
                out[(n * OC + oc) * HW + rem] = acc[t][s][i] + bias[oc];
            }
        }
    }
}

extern "C" void kernel_launch(void* const* d_in, const int* in_sizes, int n_in,
                              void* d_out, int out_size, void* d_ws, size_t ws_size,
                              hipStream_t stream) {
    const float* x    = (const float*)d_in[0];
    const float* wgt  = (const float*)d_in[1];
    const float* bias = (const float*)d_in[2];
    float* out        = (float*)d_out;

    const int nblocks = NSP / SPB;  // 1568
    conv3x3_wmma_f32<<<dim3(nblocks), dim3(256), 0, stream>>>(x, wgt, bias, out);
}